// GINEGCN_4174708212102
// MI455X (gfx1250) — compile-verified
//
#include <hip/hip_runtime.h>
#include <hip/hip_bf16.h>

// ---------------- problem constants (match reference) ----------------
#define N_NODES 50000
#define N_EDGES 800000
#define FIN     64
#define EDIM    16
#define HDIM    128
#define NGRAPH  256
#define BN_EPS  1e-5f

typedef float v2f __attribute__((ext_vector_type(2)));
typedef float v8f __attribute__((ext_vector_type(8)));

// ---------------------------------------------------------------------
// Edge phase: one wave32 per edge.
//   e_c   = eb[c] + sum_k edge_attr[e][k] * ew[k][c]
//   msg_c = relu(x[src][c] + e_c)
//   atomicAdd(aggr[dst][c], msg_c)
// Lanes cover channels c = lane + 32*j. Weight (<=8KB) stays hot in cache.
// ---------------------------------------------------------------------
template <int CIN>
__global__ __launch_bounds__(256)
void edge_gather_scatter(const float* __restrict__ edge_attr,
                         const int*   __restrict__ edge_index,
                         const float* __restrict__ ew,
                         const float* __restrict__ ebv,
                         const float* __restrict__ xin,
                         float*       __restrict__ aggr)
{
    const int wave = (blockIdx.x * blockDim.x + threadIdx.x) >> 5;
    const int lane = threadIdx.x & 31;
    if (wave >= N_EDGES) return;                  // uniform per wave

    const int src = edge_index[wave];
    const int dst = edge_index[N_EDGES + wave];

    constexpr int J = CIN / 32;
    float acc[J];
#pragma unroll
    for (int j = 0; j < J; ++j) acc[j] = ebv[lane + 32 * j];

    const float* ea = edge_attr + (size_t)wave * EDIM;
#pragma unroll
    for (int k = 0; k < EDIM; ++k) {
        const float e = ea[k];                    // broadcast load
        const float* wr = ew + (size_t)k * CIN;
#pragma unroll
        for (int j = 0; j < J; ++j)
            acc[j] = fmaf(e, wr[lane + 32 * j], acc[j]);
    }

    const float* xr = xin  + (size_t)src * CIN;
    float*       ar = aggr + (size_t)dst * CIN;
#pragma unroll
    for (int j = 0; j < J; ++j) {
        float m = acc[j] + xr[lane + 32 * j];
        m = m > 0.f ? m : 0.f;
        unsafeAtomicAdd(&ar[lane + 32 * j], m);   // global_atomic_add_f32
    }
}

// ---------------------------------------------------------------------
// Node MLP via V_WMMA_F32_16X16X4_F32.
//   h = relu((aggr + x) @ W1 + b1) @ W2 + b2     (16-row strip per wave)
// Also accumulates BN column sum / sumsq with atomics.
// ---------------------------------------------------------------------
#define LDSTRIDE 132   // 128 + 4 floats: conflict-free column reads

template <int CIN>
__global__ __launch_bounds__(128)
void node_mlp(const float* __restrict__ aggr, const float* __restrict__ xin,
              const float* __restrict__ w1,   const float* __restrict__ b1,
              const float* __restrict__ w2,   const float* __restrict__ b2,
              float* __restrict__ hout,
              float* __restrict__ colsum, float* __restrict__ colsumsq)
{
    __shared__ float lds[4][16 * LDSTRIDE];      // 33.8 KB, per-wave strips

    const int waveId = threadIdx.x >> 5;
    const int lane   = threadIdx.x & 31;
    const int strip  = blockIdx.x * 4 + waveId;
    if (strip * 16 >= N_NODES) return;           // uniform per wave
    const int row0 = strip * 16;

    float* buf = lds[waveId];

    // Stage A = aggr + x into LDS with coalesced global reads.
    for (int idx = lane; idx < 16 * CIN; idx += 32) {
        const int m = idx / CIN, k = idx % CIN;  // CIN is a power of two
        const size_t g = (size_t)(row0 + m) * CIN + k;
        buf[m * LDSTRIDE + k] = aggr[g] + xin[g];
    }

    const int mrow  = lane & 15;                 // A: M index
    const int khalf = (lane >> 4) * 2;           // A/B: K sub-offset
    const int ncol  = lane & 15;                 // B/C/D: N index
    const int mhi   = (lane >> 4) * 8;           // C/D: M offset

    // ---- GEMM1: [16,CIN] @ W1[CIN,128] -> 8 C-tiles -------------------
    v8f acc[8];
#pragma unroll
    for (int ct = 0; ct < 8; ++ct) acc[ct] = (v8f){0,0,0,0,0,0,0,0};

    for (int k0 = 0; k0 < CIN; k0 += 4) {
        v2f a;
        a.x = buf[mrow * LDSTRIDE + k0 + khalf];
        a.y = buf[mrow * LDSTRIDE + k0 + khalf + 1];
#pragma unroll
        for (int ct = 0; ct < 8; ++ct) {
            const float* wp = w1 + (size_t)(k0 + khalf) * HDIM + ct * 16 + ncol;
            v2f b; b.x = wp[0]; b.y = wp[HDIM];
            acc[ct] = __builtin_amdgcn_wmma_f32_16x16x4_f32(
                          false, a, false, b, (short)0, acc[ct], false, false);
        }
    }

    // bias + relu, intermediate back into the same per-wave LDS strip
#pragma unroll
    for (int ct = 0; ct < 8; ++ct) {
        const float bias = b1[ct * 16 + ncol];
#pragma unroll
        for (int r = 0; r < 8; ++r) {
            float v = acc[ct][r] + bias;
            v = v > 0.f ? v : 0.f;
            buf[(r + mhi) * LDSTRIDE + ct * 16 + ncol] = v;
        }
    }

    // ---- GEMM2: [16,128] @ W2[128,128] -------------------------------
    v8f acc2[8];
#pragma unroll
    for (int ct = 0; ct < 8; ++ct) acc2[ct] = (v8f){0,0,0,0,0,0,0,0};

    for (int k0 = 0; k0 < HDIM; k0 += 4) {
        v2f a;
        a.x = buf[mrow * LDSTRIDE + k0 + khalf];
        a.y = buf[mrow * LDSTRIDE + k0 + khalf + 1];
#pragma unroll
        for (int ct = 0; ct < 8; ++ct) {
            const float* wp = w2 + (size_t)(k0 + khalf) * HDIM + ct * 16 + ncol;
            v2f b; b.x = wp[0]; b.y = wp[HDIM];
            acc2[ct] = __builtin_amdgcn_wmma_f32_16x16x4_f32(
                           false, a, false, b, (short)0, acc2[ct], false, false);
        }
    }

    // bias, write h, accumulate BN stats
#pragma unroll
    for (int ct = 0; ct < 8; ++ct) {
        const float bias = b2[ct * 16 + ncol];
        float s = 0.f, s2 = 0.f;
#pragma unroll
        for (int r = 0; r < 8; ++r) {
            const float v = acc2[ct][r] + bias;
            hout[(size_t)(row0 + r + mhi) * HDIM + ct * 16 + ncol] = v;
            s += v; s2 += v * v;
        }
        unsafeAtomicAdd(&colsum[ct * 16 + ncol], s);
        unsafeAtomicAdd(&colsumsq[ct * 16 + ncol], s2);
    }
}

// ---------------------------------------------------------------------
__global__ void bn_finalize(const float* __restrict__ colsum,
                            const float* __restrict__ colsumsq,
                            const float* __restrict__ g,
                            const float* __restrict__ b,
                            float* __restrict__ scale,
                            float* __restrict__ shift)
{
    const int c = threadIdx.x;
    if (c < HDIM) {
        const float inv = 1.0f / (float)N_NODES;
        const float mu  = colsum[c] * inv;
        const float var = colsumsq[c] * inv - mu * mu;
        const float sc  = g[c] * rsqrtf(var + BN_EPS);
        scale[c] = sc;
        shift[c] = b[c] - mu * sc;
    }
}

__global__ void bn_apply(const float* __restrict__ hin,
                         const float* __restrict__ scale,
                         const float* __restrict__ shift,
                         float* __restrict__ xout)
{
    const size_t i = (size_t)blockIdx.x * blockDim.x + threadIdx.x;
    if (i < (size_t)N_NODES * HDIM) {
        const int c = (int)(i & (HDIM - 1));
        float v = fmaf(hin[i], scale[c], shift[c]);
        xout[i] = v > 0.f ? v : 0.f;
    }
}

__global__ void zero_f32(float* __restrict__ p, size_t n)
{
    size_t i  = (size_t)blockIdx.x * blockDim.x + threadIdx.x;
    const size_t st = (size_t)gridDim.x * blockDim.x;
    for (; i < n; i += st) p[i] = 0.f;
}

// wave per node: segment-sum into [G,H] pool + counts
__global__ __launch_bounds__(256)
void pool_accum(const float* __restrict__ x, const int* __restrict__ batch,
                float* __restrict__ pool, float* __restrict__ counts)
{
    const int node = (blockIdx.x * blockDim.x + threadIdx.x) >> 5;
    const int lane = threadIdx.x & 31;
    if (node >= N_NODES) return;
    const int g = batch[node];
#pragma unroll
    for (int j = 0; j < 4; ++j)
        unsafeAtomicAdd(&pool[(size_t)g * HDIM + lane + 32 * j],
                        x[(size_t)node * HDIM + lane + 32 * j]);
    if (lane == 0) unsafeAtomicAdd(&counts[g], 1.0f);
}

__global__ void final_lin(const float* __restrict__ pool,
                          const float* __restrict__ counts,
                          const float* __restrict__ lw,
                          const float* __restrict__ lb,
                          float* __restrict__ out)
{
    const int g = blockIdx.x * blockDim.x + threadIdx.x;
    if (g < NGRAPH) {
        float s = 0.f;
        for (int c = 0; c < HDIM; ++c) s += pool[(size_t)g * HDIM + c] * lw[c];
        float cnt = counts[g];
        cnt = cnt > 1.f ? cnt : 1.f;
        out[g] = s / cnt + lb[0];
    }
}

// ---------------------------------------------------------------------
extern "C" void kernel_launch(void* const* d_in, const int* in_sizes, int n_in,
                              void* d_out, int out_size, void* d_ws, size_t ws_size,
                              hipStream_t stream)
{
    const float* x_in      = (const float*)d_in[0];
    const float* edge_attr = (const float*)d_in[1];
    const int*   edge_idx  = (const int*)  d_in[2];
    const int*   batch     = (const int*)  d_in[3];
    // per-layer params: e_w, e_b, w1, b1, w2, b2, bn_g, bn_b  (8 each)
    const float* P[3][8];
    for (int l = 0; l < 3; ++l)
        for (int j = 0; j < 8; ++j)
            P[l][j] = (const float*)d_in[4 + l * 8 + j];
    const float* lin_w = (const float*)d_in[28];
    const float* lin_b = (const float*)d_in[29];
    float* out = (float*)d_out;

    // workspace carve-up (~103 MB of f32)
    float* ws       = (float*)d_ws;
    float* aggr     = ws;                                   // N*128
    float* hbuf     = aggr + (size_t)N_NODES * HDIM;        // N*128
    float* xA       = hbuf + (size_t)N_NODES * HDIM;        // N*128
    float* xB       = xA   + (size_t)N_NODES * HDIM;        // N*128
    float* colsum   = xB   + (size_t)N_NODES * HDIM;        // 128
    float* colsumsq = colsum + HDIM;                        // 128
    float* scale    = colsumsq + HDIM;                      // 128
    float* shift    = scale + HDIM;                         // 128
    float* pool     = shift + HDIM;                         // G*128
    float* counts   = pool + (size_t)NGRAPH * HDIM;         // G

    const int edgeBlocks = N_EDGES / 8;                     // wave per edge, 8 waves/blk
    const int mlpBlocks  = (N_NODES / 16 + 3) / 4;          // 16-row strip per wave
    const int bnBlocks   = (int)(((size_t)N_NODES * HDIM + 255) / 256);

    // ---------------- layer 1 (CIN = 64) ----------------
    zero_f32<<<1024, 256, 0, stream>>>(aggr, (size_t)N_NODES * FIN);
    zero_f32<<<1, 256, 0, stream>>>(colsum, 2 * HDIM);
    edge_gather_scatter<FIN><<<edgeBlocks, 256, 0, stream>>>(
        edge_attr, edge_idx, P[0][0], P[0][1], x_in, aggr);
    node_mlp<FIN><<<mlpBlocks, 128, 0, stream>>>(
        aggr, x_in, P[0][2], P[0][3], P[0][4], P[0][5], hbuf, colsum, colsumsq);
    bn_finalize<<<1, 128, 0, stream>>>(colsum, colsumsq, P[0][6], P[0][7], scale, shift);
    bn_apply<<<bnBlocks, 256, 0, stream>>>(hbuf, scale, shift, xA);

    // ---------------- layer 2 (CIN = 128) ----------------
    zero_f32<<<1024, 256, 0, stream>>>(aggr, (size_t)N_NODES * HDIM);
    zero_f32<<<1, 256, 0, stream>>>(colsum, 2 * HDIM);
    edge_gather_scatter<HDIM><<<edgeBlocks, 256, 0, stream>>>(
        edge_attr, edge_idx, P[1][0], P[1][1], xA, aggr);
    node_mlp<HDIM><<<mlpBlocks, 128, 0, stream>>>(
        aggr, xA, P[1][2], P[1][3], P[1][4], P[1][5], hbuf, colsum, colsumsq);
    bn_finalize<<<1, 128, 0, stream>>>(colsum, colsumsq, P[1][6], P[1][7], scale, shift);
    bn_apply<<<bnBlocks, 256, 0, stream>>>(hbuf, scale, shift, xB);

    // ---------------- layer 3 (CIN = 128) ----------------
    zero_f32<<<1024, 256, 0, stream>>>(aggr, (size_t)N_NODES * HDIM);
    zero_f32<<<1, 256, 0, stream>>>(colsum, 2 * HDIM);
    edge_gather_scatter<HDIM><<<edgeBlocks, 256, 0, stream>>>(
        edge_attr, edge_idx, P[2][0], P[2][1], xB, aggr);
    node_mlp<HDIM><<<mlpBlocks, 128, 0, stream>>>(
        aggr, xB, P[2][2], P[2][3], P[2][4], P[2][5], hbuf, colsum, colsumsq);
    bn_finalize<<<1, 128, 0, stream>>>(colsum, colsumsq, P[2][6], P[2][7], scale, shift);
    bn_apply<<<bnBlocks, 256, 0, stream>>>(hbuf, scale, shift, xA);

    // ---------------- global mean pool + linear ----------------
    zero_f32<<<64, 256, 0, stream>>>(pool, (size_t)NGRAPH * HDIM + NGRAPH);
    pool_accum<<<(N_NODES + 7) / 8, 256, 0, stream>>>(xA, batch, pool, counts);
    final_lin<<<1, 256, 0, stream>>>(pool, counts, lin_w, lin_b, out);
}